// ItemInfluenceEmbedding_9216999817725
// MI455X (gfx1250) — compile-verified
//
#include <hip/hip_runtime.h>

typedef float v2f __attribute__((ext_vector_type(2)));
typedef float v8f __attribute__((ext_vector_type(8)));

#define D 16
#define ATTN_SLOPE 0.2f
#define ACT_SLOPE  0.01f

// ---- order-preserving float<->uint encoding for atomic segment-max ----
__device__ __forceinline__ unsigned enc_f32(float f) {
    unsigned u = __float_as_uint(f);
    return (u & 0x80000000u) ? ~u : (u | 0x80000000u);
}
__device__ __forceinline__ float dec_f32(unsigned u) {
    unsigned b = (u & 0x80000000u) ? (u ^ 0x80000000u) : ~u;
    return __uint_as_float(b);
}

// ---- node transform: feat_src = h@Wsrc+bsrc, feat_dst = h@Wdst+bdst ----
// One wave computes one 16-row tile using V_WMMA_F32_16X16X4_F32 (4 K-steps),
// sharing the A operand between the two GEMMs.
__global__ void node_transform_wmma(const float* __restrict__ h,
                                    const float* __restrict__ Wsrc,
                                    const float* __restrict__ bsrc,
                                    const float* __restrict__ Wdst,
                                    const float* __restrict__ bdst,
                                    float* __restrict__ fs,
                                    float* __restrict__ fd,
                                    int ntiles) {
    const int lane = threadIdx.x & 31;
    const int wave = threadIdx.x >> 5;
    const int tile = blockIdx.x * (blockDim.x >> 5) + wave;
    if (tile >= ntiles) return;          // wave-uniform branch, EXEC stays full
    const int row0 = tile * 16;
    const int col  = lane & 15;
    const int hi   = lane >> 4;

    // B operand (4x16 f32): VGPR v holds row K = v + 2*hi, col = lane&15
    v2f bs[4], bd[4];
#pragma unroll
    for (int kb = 0; kb < 4; ++kb) {
        const int k0 = kb * 4 + 2 * hi;
        bs[kb].x = Wsrc[k0 * D + col];
        bs[kb].y = Wsrc[(k0 + 1) * D + col];
        bd[kb].x = Wdst[k0 * D + col];
        bd[kb].y = Wdst[(k0 + 1) * D + col];
    }

    v8f accS = {};
    v8f accD = {};
#pragma unroll
    for (int kb = 0; kb < 4; ++kb) {
        const int k0 = kb * 4 + 2 * hi;
        // A operand (16x4 f32): lane holds row M = lane&15, K = 2*hi + {0,1}
        const float* ap = h + (size_t)(row0 + col) * D + k0;
        v2f a;
        a.x = ap[0];
        a.y = ap[1];
        accS = __builtin_amdgcn_wmma_f32_16x16x4_f32(false, a, false, bs[kb],
                                                     (short)0, accS, false, false);
        accD = __builtin_amdgcn_wmma_f32_16x16x4_f32(false, a, false, bd[kb],
                                                     (short)0, accD, false, false);
    }

    const float biasS = bsrc[col];
    const float biasD = bdst[col];
    // C/D layout: VGPR v -> row = row0 + v + 8*hi, col = lane&15
#pragma unroll
    for (int v = 0; v < 8; ++v) {
        const int row = row0 + v + 8 * hi;
        fs[(size_t)row * D + col] = accS[v] + biasS;
        fd[(size_t)row * D + col] = accD[v] + biasD;
    }
}

// ---- per-layer init: rst=0, denom=0, smax=enc(-inf-ish)=0 ----
__global__ void init_layer(float* __restrict__ rst, float* __restrict__ denom,
                           unsigned* __restrict__ smax, int n) {
    const int i = blockIdx.x * blockDim.x + threadIdx.x;
    if (i < n * D) rst[i] = 0.0f;
    if (i < n) {
        denom[i] = 0.0f;
        smax[i]  = 0u;   // below enc() of any real float -> identity for max
    }
}

// ---- edge pass 1: score[e] = leaky(el+ed, .2) . attn ; segment max ----
__global__ void edge_score_max(const float* __restrict__ fs,
                               const float* __restrict__ fd,
                               const int* __restrict__ src,
                               const int* __restrict__ dst,
                               const float* __restrict__ attn,
                               float* __restrict__ score,
                               unsigned* __restrict__ smax, int E_) {
    const int e = blockIdx.x * blockDim.x + threadIdx.x;
    if (e >= E_) return;
    const int s = src[e], d = dst[e];
    const float4* a4 = (const float4*)(fs + (size_t)s * D);
    const float4* b4 = (const float4*)(fd + (size_t)d * D);
    const float4* w4 = (const float4*)attn;
    float sc = 0.0f;
#pragma unroll
    for (int j = 0; j < 4; ++j) {
        const float4 a = a4[j];
        const float4 b = b4[j];
        const float4 w = w4[j];
        float vx = a.x + b.x; vx = vx > 0.0f ? vx : ATTN_SLOPE * vx;
        float vy = a.y + b.y; vy = vy > 0.0f ? vy : ATTN_SLOPE * vy;
        float vz = a.z + b.z; vz = vz > 0.0f ? vz : ATTN_SLOPE * vz;
        float vw = a.w + b.w; vw = vw > 0.0f ? vw : ATTN_SLOPE * vw;
        sc += vx * w.x + vy * w.y + vz * w.z + vw * w.w;
    }
    score[e] = sc;
    atomicMax(smax + d, enc_f32(sc));
}

// ---- edge pass 2: ex = exp(score - smax[dst]); denom += ex (in-place) ----
__global__ void edge_exp_denom(const int* __restrict__ dst,
                               const unsigned* __restrict__ smax,
                               float* __restrict__ score_ex,
                               float* __restrict__ denom, int E_) {
    const int e = blockIdx.x * blockDim.x + threadIdx.x;
    if (e >= E_) return;
    const int d = dst[e];
    const float ex = __expf(score_ex[e] - dec_f32(smax[d]));
    score_ex[e] = ex;
    unsafeAtomicAdd(denom + d, ex);
}

// ---- edge pass 3: rst[dst] += (ex/denom[dst]) * feat_src[src] ----
__global__ void edge_scatter(const float* __restrict__ fs,
                             const int* __restrict__ src,
                             const int* __restrict__ dst,
                             const float* __restrict__ exb,
                             const float* __restrict__ denom,
                             float* __restrict__ rst, int E_) {
    const int e = blockIdx.x * blockDim.x + threadIdx.x;
    if (e >= E_) return;
    const int s = src[e], d = dst[e];
    const float alpha = exb[e] / denom[d];
    const float* el = fs + (size_t)s * D;
    float* out = rst + (size_t)d * D;
#pragma unroll
    for (int j = 0; j < D; ++j) unsafeAtomicAdd(out + j, alpha * el[j]);
}

// ---- node finalize: leaky_relu(rst, .01) -> layer output ----
__global__ void node_act(const float* __restrict__ rst, float* __restrict__ hout,
                         int n16) {
    const int i = blockIdx.x * blockDim.x + threadIdx.x;
    if (i >= n16) return;
    const float v = rst[i];
    hout[i] = v > 0.0f ? v : ACT_SLOPE * v;
}

extern "C" void kernel_launch(void* const* d_in, const int* in_sizes, int n_in,
                              void* d_out, int out_size, void* d_ws, size_t ws_size,
                              hipStream_t stream) {
    const float* emb   = (const float*)d_in[0];
    const int*   src1  = (const int*)d_in[1];
    const int*   dst1  = (const int*)d_in[2];
    const int*   src2  = (const int*)d_in[3];
    const int*   dst2  = (const int*)d_in[4];
    const float* Wsrc1 = (const float*)d_in[5];
    const float* bsrc1 = (const float*)d_in[6];
    const float* Wdst1 = (const float*)d_in[7];
    const float* bdst1 = (const float*)d_in[8];
    const float* attn1 = (const float*)d_in[9];
    const float* Wsrc2 = (const float*)d_in[10];
    const float* bsrc2 = (const float*)d_in[11];
    const float* Wdst2 = (const float*)d_in[12];
    const float* bdst2 = (const float*)d_in[13];
    const float* attn2 = (const float*)d_in[14];

    const int Nn = in_sizes[0] / D;
    const int Ee = in_sizes[1];

    float* ws = (float*)d_ws;
    size_t o = 0;
    float* fs   = ws + o; o += (size_t)Nn * D;
    float* fd   = ws + o; o += (size_t)Nn * D;
    float* exb  = ws + o; o += (size_t)Ee;       // score, overwritten by ex
    unsigned* smax = (unsigned*)(ws + o); o += (size_t)Nn;
    float* denom = ws + o; o += (size_t)Nn;
    float* rst   = ws + o; o += (size_t)Nn * D;
    float* h2    = ws + o; o += (size_t)Nn * D;
    (void)ws_size;

    const int ntiles = (Nn + 15) / 16;
    const dim3 tfmGrid((ntiles + 7) / 8), tfmBlk(256);       // 8 waves/block
    const dim3 edgeGrid((Ee + 255) / 256), edgeBlk(256);
    const dim3 nodeGrid((Nn * D + 255) / 256), nodeBlk(256);

    // ---------------- layer 1 ----------------
    init_layer<<<nodeGrid, nodeBlk, 0, stream>>>(rst, denom, smax, Nn);
    node_transform_wmma<<<tfmGrid, tfmBlk, 0, stream>>>(emb, Wsrc1, bsrc1, Wdst1,
                                                        bdst1, fs, fd, ntiles);
    edge_score_max<<<edgeGrid, edgeBlk, 0, stream>>>(fs, fd, src1, dst1, attn1,
                                                     exb, smax, Ee);
    edge_exp_denom<<<edgeGrid, edgeBlk, 0, stream>>>(dst1, smax, exb, denom, Ee);
    edge_scatter<<<edgeGrid, edgeBlk, 0, stream>>>(fs, src1, dst1, exb, denom,
                                                   rst, Ee);
    node_act<<<nodeGrid, nodeBlk, 0, stream>>>(rst, h2, Nn * D);

    // ---------------- layer 2 ----------------
    init_layer<<<nodeGrid, nodeBlk, 0, stream>>>(rst, denom, smax, Nn);
    node_transform_wmma<<<tfmGrid, tfmBlk, 0, stream>>>(h2, Wsrc2, bsrc2, Wdst2,
                                                        bdst2, fs, fd, ntiles);
    edge_score_max<<<edgeGrid, edgeBlk, 0, stream>>>(fs, fd, src2, dst2, attn2,
                                                     exb, smax, Ee);
    edge_exp_denom<<<edgeGrid, edgeBlk, 0, stream>>>(dst2, smax, exb, denom, Ee);
    edge_scatter<<<edgeGrid, edgeBlk, 0, stream>>>(fs, src2, dst2, exb, denom,
                                                   rst, Ee);
    node_act<<<nodeGrid, nodeBlk, 0, stream>>>(rst, (float*)d_out, Nn * D);
}